// FirstFrameBiasedEMA_19490561590049
// MI455X (gfx1250) — compile-verified
//
#include <hip/hip_runtime.h>

// First-frame-biased EMA, y[t] = a*y[t-1] + (1-a)*x[t], y[0]=x[0]
// Chunk-parallel scan with warm-up tail; 16-step blocks via
// V_WMMA_F32_16X16X4_F32 with a constant lower-triangular coefficient matrix.
// 4-buffer register pipeline, immediate-offset loads AND stores, VALU-only
// (permlanex16) carry hand-off, non-temporal result stores, and
// counter-free global_prefetch_b8 software prefetch 8 tiles ahead.

typedef float v2f __attribute__((ext_vector_type(2)));
typedef float v8f __attribute__((ext_vector_type(8)));

#define EMA_ALPHA 0.9f
#define EMA_BETA  0.1f

// alpha^n for n = 0..15
__device__ __constant__ float c_apow[16] = {
    1.0f,            0.9f,            0.81f,           0.729f,
    0.6561f,         0.59049f,        0.531441f,       0.4782969f,
    0.43046721f,     0.387420489f,    0.3486784401f,   0.31381059609f,
    0.282429536481f, 0.2541865828329f,0.22876792454961f,0.205891132094649f
};

struct Btile { v2f b[4]; };

// Move D row 15 (lanes 16-31) into lanes 0-15: cross-half identity permute.
__device__ __forceinline__ float carry_xchg(float v) {
#if __has_builtin(__builtin_amdgcn_permlanex16)
    int s = __builtin_bit_cast(int, v);
    int p = __builtin_amdgcn_permlanex16(s, s, 0x76543210, 0xfedcba98,
                                         false, false);
    return __builtin_bit_cast(float, p);
#else
    return __shfl(v, (threadIdx.x & 15) + 16, 32);
#endif
}

__global__ __launch_bounds__(256)
void ema_wmma_kernel(const float* __restrict__ x, float* __restrict__ y) {
    constexpr int T = 4096, C = 512;
    constexpr int L = 512;   // timesteps per chunk
    constexpr int W = 256;   // warm-up steps (alpha^256 ~ 2e-12, < fp32 eps)
    constexpr int PFD = 8;   // prefetch distance in 16-step tiles

    const int lane = threadIdx.x & 31;
    const int wv   = threadIdx.x >> 5;     // 8 waves per block
    const int li   = lane & 15;            // channel within 16-group / row M
    const int h    = lane >> 4;            // half-wave select

    const int c0    = blockIdx.x * 128 + wv * 16;  // channel base (C/128 = 4)
    const int b     = blockIdx.z;                  // batch
    const int chunk = blockIdx.y;                  // T/L = 8 chunks
    const int t0    = chunk * L;
    const int tw    = (chunk == 0) ? 0 : (t0 - W);
    const int nT    = ((t0 + L) - tw) >> 4;        // 16-step tiles (32 or 48)

    // Per-lane base pointers with the half-wave row bias pre-folded, so every
    // load/store in a tile becomes base[const*C] -> immediate 24-bit offset.
    const float* __restrict__ xph =
        x + (size_t)b * T * C + c0 + li + (size_t)(2 * h) * C;  // B rows K=2h+...
    float* __restrict__ yph =
        y + (size_t)b * T * C + c0 + li + (size_t)(8 * h) * C;  // D rows M=8h+r

    // Prefetch pointer: lane L covers row (L%16) of a tile at column block c0
    // (64B contiguous per row -> full line pulled; upper half dedups in cache).
    const float* __restrict__ xpf =
        x + (size_t)b * T * C + c0 + (size_t)li * C;

    // Constant A matrix (16x16 lower-triangular, M[i][k] = beta*alpha^(i-k)),
    // four 16x4 K-slabs in WMMA f32 A layout (VGPR0: K=4j+2h, VGPR1: K=4j+2h+1).
    v2f A[4];
#pragma unroll
    for (int j = 0; j < 4; ++j) {
        const int k0 = 4 * j + 2 * h;
        const float a0 = (k0     <= li) ? EMA_BETA * c_apow[li - k0]       : 0.0f;
        const float a1 = (k0 + 1 <= li) ? EMA_BETA * c_apow[li - (k0 + 1)] : 0.0f;
        A[j] = (v2f){a0, a1};
    }
    // Carry matrix: single K=0 column of alpha^(i+1) (lanes 0-15, VGPR0)
    const v2f Ac = (v2f){(h == 0) ? (EMA_ALPHA * c_apow[li]) : 0.0f, 0.0f};

    // Fresh-start carry: y_prev = x[tw]. Since 1-beta == alpha this yields
    // y[tw] = x[tw] exactly (covers the y[0] = x[0] first-frame rule).
    float yprev = xph[0 - (size_t)(2 * h) * C + (size_t)tw * C];

    // Tile base pointer for tile index idx (uniform offset -> SALU + 1 v_add)
    auto tilePtr = [&](int idx) -> const float* {
        return xph + (size_t)(tw + idx * 16) * C;
    };
    // 8 loads, all immediate offsets off one base
    auto loadT = [&](const float* base) {
        Btile t;
#pragma unroll
        for (int j = 0; j < 4; ++j)
            t.b[j] = (v2f){ base[(size_t)(4 * j) * C],
                            base[(size_t)(4 * j + 1) * C] };
        return t;
    };

    // Preload tiles 0..3 into the rotation buffers
    Btile buf[4];
#pragma unroll
    for (int k = 0; k < 4; ++k)
        buf[k] = loadT(tilePtr(k));   // nT >= 32, always valid

    for (int ti = 0; ti < nT; ti += 4) {
#pragma unroll
        for (int k = 0; k < 4; ++k) {
            const int t_idx = ti + k;
            const int tt    = tw + t_idx * 16;

            // Counter-free line prefetch PFD tiles ahead (global_prefetch_b8)
            const int pf_idx = min(t_idx + PFD, nT - 1);
            __builtin_prefetch(xpf + (size_t)(tw + pf_idx * 16) * C, 0, 3);

            // D = carry + M * X  (5 WMMAs)
            v8f d = {};
            const v2f Bc = (v2f){(h == 0) ? yprev : 0.0f, 0.0f};
            d = __builtin_amdgcn_wmma_f32_16x16x4_f32(false, Ac, false, Bc,
                                                      (short)0, d, false, false);
#pragma unroll
            for (int j = 0; j < 4; ++j)
                d = __builtin_amdgcn_wmma_f32_16x16x4_f32(false, A[j], false,
                                                          buf[k].b[j],
                                                          (short)0, d, false, false);

            // New carry = y[tt+15][c] (D VGPR7, lanes 16-31 -> lanes 0-15)
            yprev = carry_xchg(d[7]);

            // Refill this buffer with tile t_idx+4 (clamped; WAR after consume)
            buf[k] = loadT(tilePtr(min(t_idx + 4, nT - 1)));

            if (tt >= t0) {                       // skip warm-up outputs
                float* __restrict__ yb = yph + (size_t)tt * C;
#pragma unroll
                for (int r = 0; r < 8; ++r)       // immediate offsets r*2048
                    __builtin_nontemporal_store(d[r], &yb[(size_t)r * C]);
            }
        }
    }
}

extern "C" void kernel_launch(void* const* d_in, const int* in_sizes, int n_in,
                              void* d_out, int out_size, void* d_ws, size_t ws_size,
                              hipStream_t stream) {
    (void)in_sizes; (void)n_in; (void)out_size; (void)d_ws; (void)ws_size;
    const float* x = (const float*)d_in[0];
    float*       y = (float*)d_out;
    // grid: (C/128 channel groups, T/L chunks, B batches), 256 threads = 8 waves
    dim3 grid(4, 8, 16);
    ema_wmma_kernel<<<grid, 256, 0, stream>>>(x, y);
}